// QTLayer_79396765434347
// MI455X (gfx1250) — compile-verified
//
#include <hip/hip_runtime.h>

#ifndef __has_builtin
#define __has_builtin(x) 0
#endif

// Problem constants (fixed by the reference).
constexpr int kB      = 1 << 20;
constexpr int kNState = 256;
constexpr int kNAct   = 64;
constexpr int kR      = 8;

// Launch shape: 512 blocks x 256 threads x 8 elements/thread == 2^20.
constexpr int kThreads = 256;
constexpr int kEPT     = 8;
constexpr int kBlocks  = kB / (kThreads * kEPT);

// Skewed LDS pitches (floats). All are multiples of 4 -> float4 aligned.
// c1/c2: [s][r][j], pitch 68 per s-slice  -> b128 bank group = (s + 2r) mod 16.
// c0   : [s][j],    pitch 12 per row      -> bank group = 3s mod 16.
// c3   : [a][r] (transposed), pitch 12    -> bank group = 3a mod 16.
constexpr int kC1Pitch = 68;
constexpr int kC0Pitch = 12;
constexpr int kC3Pitch = 12;

#if defined(__gfx1250__) &&                                             \
    __has_builtin(__builtin_amdgcn_global_load_async_to_lds_b128) &&    \
    __has_builtin(__builtin_amdgcn_global_load_async_to_lds_b32) &&     \
    __has_builtin(__builtin_amdgcn_s_wait_asynccnt)
#define USE_ASYNC_LDS 1
#else
#define USE_ASYNC_LDS 0
#endif

#if USE_ASYNC_LDS
typedef int v4i __attribute__((vector_size(16)));
typedef __attribute__((address_space(1))) v4i* gptr_b128;
typedef __attribute__((address_space(3))) v4i* lptr_b128;
typedef __attribute__((address_space(1))) int* gptr_b32;
typedef __attribute__((address_space(3))) int* lptr_b32;

__device__ __forceinline__ void async_b128(const void* gsrc, void* ldst) {
  __builtin_amdgcn_global_load_async_to_lds_b128((gptr_b128)gsrc,
                                                 (lptr_b128)ldst, 0, 0);
}
__device__ __forceinline__ void async_b32(const void* gsrc, void* ldst) {
  __builtin_amdgcn_global_load_async_to_lds_b32((gptr_b32)gsrc, (lptr_b32)ldst,
                                                0, 0);
}
#endif

// Two-phase 8x8 matvec: burst all 16 ds_load_b128 first (single dscnt wait),
// then 64 FMAs (backend packs them into v_pk_fma_f32 / dual-issue forms).
__device__ __forceinline__ void matvec8(const float* __restrict__ lds, int s,
                                        const float v[8], float u[8]) {
  float4 m[16];
  const int base = s * kC1Pitch;
#pragma unroll
  for (int r = 0; r < 8; ++r) {
    m[2 * r + 0] = *reinterpret_cast<const float4*>(lds + base + r * 8);
    m[2 * r + 1] = *reinterpret_cast<const float4*>(lds + base + r * 8 + 4);
  }
#pragma unroll
  for (int j = 0; j < 8; ++j) u[j] = 0.0f;
#pragma unroll
  for (int r = 0; r < 8; ++r) {
    const float vr = v[r];
    u[0] = fmaf(vr, m[2 * r].x, u[0]);
    u[1] = fmaf(vr, m[2 * r].y, u[1]);
    u[2] = fmaf(vr, m[2 * r].z, u[2]);
    u[3] = fmaf(vr, m[2 * r].w, u[3]);
    u[4] = fmaf(vr, m[2 * r + 1].x, u[4]);
    u[5] = fmaf(vr, m[2 * r + 1].y, u[5]);
    u[6] = fmaf(vr, m[2 * r + 1].z, u[6]);
    u[7] = fmaf(vr, m[2 * r + 1].w, u[7]);
  }
}

__global__ __launch_bounds__(kThreads) void qtt_kernel(
    const int* __restrict__ states,    // (B,3) int
    const int* __restrict__ actions,   // (B,)  int
    const float* __restrict__ core0,   // 256*8
    const float* __restrict__ core1,   // 8*256*8
    const float* __restrict__ core2,   // 8*256*8
    const float* __restrict__ core3,   // 8*64
    float* __restrict__ out) {
  __shared__ __align__(16) float lds_c1[kNState * kC1Pitch];  // 17408 f
  __shared__ __align__(16) float lds_c2[kNState * kC1Pitch];  // 17408 f
  __shared__ __align__(16) float lds_c0[kNState * kC0Pitch];  //  3072 f
  __shared__ __align__(16) float lds_c3[kNAct * kC3Pitch];    //   768 f

  const int tid = threadIdx.x;

  // ---- Stage all four TT-cores into LDS (CDNA5 async global->LDS path) ----
#if USE_ASYNC_LDS
  {
    const float4* g1 = reinterpret_cast<const float4*>(core1);
    const float4* g2 = reinterpret_cast<const float4*>(core2);
    const float4* g0 = reinterpret_cast<const float4*>(core0);
    float4* l1 = reinterpret_cast<float4*>(lds_c1);
    float4* l2 = reinterpret_cast<float4*>(lds_c2);
    float4* l0 = reinterpret_cast<float4*>(lds_c0);
#pragma unroll
    for (int k = 0; k < 16; ++k) {
      const int i = tid + k * kThreads;  // 0..4095 float4s of a core
      const int r = i >> 9;              // global layout: r*2048 + s*8 + j
      const int rem = i & 511;
      const int s = rem >> 1, jh = rem & 1;
      const int d = s * (kC1Pitch / 4) + r * 2 + jh;  // skewed [s][r][jh]
      async_b128(g1 + i, l1 + d);
      async_b128(g2 + i, l2 + d);
    }
#pragma unroll
    for (int k = 0; k < 2; ++k) {
      const int i = tid + k * kThreads;  // 0..511
      const int s = i >> 1, jh = i & 1;
      async_b128(g0 + i, l0 + s * (kC0Pitch / 4) + jh);
      const int r = i >> 6, a = i & 63;  // transpose core3 into [a][r]
      async_b32(core3 + i, lds_c3 + a * kC3Pitch + r);
    }
    __builtin_amdgcn_s_wait_asynccnt(0);
  }
#else
  {  // synchronous fallback staging
    const float4* g1 = reinterpret_cast<const float4*>(core1);
    const float4* g2 = reinterpret_cast<const float4*>(core2);
    const float4* g0 = reinterpret_cast<const float4*>(core0);
    float4* l1 = reinterpret_cast<float4*>(lds_c1);
    float4* l2 = reinterpret_cast<float4*>(lds_c2);
    float4* l0 = reinterpret_cast<float4*>(lds_c0);
#pragma unroll
    for (int k = 0; k < 16; ++k) {
      const int i = tid + k * kThreads;
      const int r = i >> 9;
      const int rem = i & 511;
      const int s = rem >> 1, jh = rem & 1;
      const int d = s * (kC1Pitch / 4) + r * 2 + jh;
      l1[d] = g1[i];
      l2[d] = g2[i];
    }
#pragma unroll
    for (int k = 0; k < 2; ++k) {
      const int i = tid + k * kThreads;
      const int s = i >> 1, jh = i & 1;
      l0[s * (kC0Pitch / 4) + jh] = g0[i];
      const int r = i >> 6, a = i & 63;
      lds_c3[a * kC3Pitch + r] = core3[i];
    }
  }
#endif
  __syncthreads();

  // ---- Per-element TT contraction, all gathers served from LDS ----
  const int base = blockIdx.x * (kThreads * kEPT);
#pragma unroll
  for (int k = 0; k < kEPT; ++k) {
    const int b = base + k * kThreads + tid;  // coalesced index loads/stores
    const int s0 = states[3 * b + 0];
    const int s1 = states[3 * b + 1];
    const int s2 = states[3 * b + 2];
    const int a = actions[b];

    float v[8], u[8];
    {
      const float4 a0 = *reinterpret_cast<const float4*>(lds_c0 + s0 * kC0Pitch);
      const float4 a1 =
          *reinterpret_cast<const float4*>(lds_c0 + s0 * kC0Pitch + 4);
      v[0] = a0.x; v[1] = a0.y; v[2] = a0.z; v[3] = a0.w;
      v[4] = a1.x; v[5] = a1.y; v[6] = a1.z; v[7] = a1.w;
    }
    matvec8(lds_c1, s1, v, u);  // u = v . core1[:, s1, :]
    matvec8(lds_c2, s2, u, v);  // v = u . core2[:, s2, :]

    const float4 w0 = *reinterpret_cast<const float4*>(lds_c3 + a * kC3Pitch);
    const float4 w1 =
        *reinterpret_cast<const float4*>(lds_c3 + a * kC3Pitch + 4);
    float acc = v[0] * w0.x;
    acc = fmaf(v[1], w0.y, acc);
    acc = fmaf(v[2], w0.z, acc);
    acc = fmaf(v[3], w0.w, acc);
    acc = fmaf(v[4], w1.x, acc);
    acc = fmaf(v[5], w1.y, acc);
    acc = fmaf(v[6], w1.z, acc);
    acc = fmaf(v[7], w1.w, acc);
    out[b] = acc;
  }
}

extern "C" void kernel_launch(void* const* d_in, const int* in_sizes, int n_in,
                              void* d_out, int out_size, void* d_ws,
                              size_t ws_size, hipStream_t stream) {
  const int* states = (const int*)d_in[0];
  const int* actions = (const int*)d_in[1];
  const float* core0 = (const float*)d_in[2];
  const float* core1 = (const float*)d_in[3];
  const float* core2 = (const float*)d_in[4];
  const float* core3 = (const float*)d_in[5];
  float* out = (float*)d_out;

  qtt_kernel<<<dim3(kBlocks), dim3(kThreads), 0, stream>>>(
      states, actions, core0, core1, core2, core3, out);
}